// ConcatPool_4904852652368
// MI455X (gfx1250) — compile-verified
//
#include <hip/hip_runtime.h>
#include <math.h>

typedef __attribute__((ext_vector_type(2))) float v2f;
typedef __attribute__((ext_vector_type(8))) float v8f;

#define D 256
#define OUTF 128
#define NFEAT 9
#define VOCAB 100
#define NLAYERS 4
#define NEG_SLOPE 0.2f

// order-preserving float -> uint encoding so atomicMax(u32) == float max
__device__ __forceinline__ unsigned fenc(float f) {
  unsigned u = __float_as_uint(f);
  return (u & 0x80000000u) ? ~u : (u | 0x80000000u);
}
__device__ __forceinline__ float fdec(unsigned u) {
  return (u & 0x80000000u) ? __uint_as_float(u ^ 0x80000000u) : __uint_as_float(~u);
}
#define ENC_NEG_INF 0x007FFFFFu  // fenc(-inf)

__global__ void fill_u32_kernel(unsigned* __restrict__ p, unsigned v, int n) {
  int i = blockIdx.x * blockDim.x + threadIdx.x;
  if (i < n) p[i] = v;
}

// h[n][d] = sum_f atom_emb[f][x[n][f]][d]
__global__ void atom_encoder_kernel(const int* __restrict__ x, const float* __restrict__ emb,
                                    float* __restrict__ h, int N) {
  int n = blockIdx.x;
  int d = threadIdx.x;
  float acc = 0.f;
#pragma unroll
  for (int f = 0; f < NFEAT; ++f) {
    int v = x[n * NFEAT + f];
    acc += emb[((size_t)(f * VOCAB + v)) * D + d];
  }
  h[(size_t)n * D + d] = acc;
}

// C[M,Ncols] = A[M,K] @ B[K,Ncols] (+ bias). One wave computes a 16x64 strip
// (4 independent accumulators -> independent back-to-back v_wmma, A reuse x4).
// M % 16 == 0, Ncols % 64 == 0, K % 4 == 0. Exact fp32 via V_WMMA_F32_16X16X4_F32.
__global__ void wmma_gemm_f32_kernel(const float* __restrict__ A, const float* __restrict__ Bm,
                                     const float* __restrict__ bias, float* __restrict__ C,
                                     int K, int Ncols) {
  int lane = threadIdx.x;           // 0..31 (wave32)
  int m    = lane & 15;             // A row within tile / B col within tile
  int kh   = lane >> 4;             // K-half selector (0: K=0,1  1: K=2,3)
  int row0 = blockIdx.x * 16;
  int col0 = blockIdx.y * 64;

  const float* arow = A + (size_t)(row0 + m) * K;
  const float* bcol = Bm + col0 + m;

  v8f acc0 = {}, acc1 = {}, acc2 = {}, acc3 = {};
  for (int k0 = 0; k0 < K; k0 += 4) {
    int ka = k0 + 2 * kh;
    v2f a = *(const v2f*)(arow + ka);               // A[m][ka], A[m][ka+1]
    const float* br0 = bcol + (size_t)ka * Ncols;   // B row ka
    const float* br1 = br0 + Ncols;                 // B row ka+1
    v2f b0, b1, b2, b3;
    b0.x = br0[0];  b0.y = br1[0];
    b1.x = br0[16]; b1.y = br1[16];
    b2.x = br0[32]; b2.y = br1[32];
    b3.x = br0[48]; b3.y = br1[48];
    acc0 = __builtin_amdgcn_wmma_f32_16x16x4_f32(false, a, false, b0, (short)0, acc0, false, false);
    acc1 = __builtin_amdgcn_wmma_f32_16x16x4_f32(false, a, false, b1, (short)0, acc1, false, false);
    acc2 = __builtin_amdgcn_wmma_f32_16x16x4_f32(false, a, false, b2, (short)0, acc2, false, false);
    acc3 = __builtin_amdgcn_wmma_f32_16x16x4_f32(false, a, false, b3, (short)0, acc3, false, false);
  }

  // D layout: VGPR j -> lanes 0-15: (M=j, N=lane); lanes 16-31: (M=j+8, N=lane-16)
  int nn = lane & 15;
  int mb = (lane < 16) ? 0 : 8;
  float* crow = C + (size_t)(row0 + mb) * Ncols + col0 + nn;
#define STORE_TILE(ACC, T)                                             \
  {                                                                    \
    float bv = bias ? bias[col0 + (T)*16 + nn] : 0.f;                  \
    _Pragma("unroll") for (int j = 0; j < 8; ++j) {                    \
      crow[(size_t)j * Ncols + (T)*16] = (ACC)[j] + bv;                \
    }                                                                  \
  }
  STORE_TILE(acc0, 0)
  STORE_TILE(acc1, 1)
  STORE_TILE(acc2, 2)
  STORE_TILE(acc3, 3)
#undef STORE_TILE
}

// s[n] = hW[n]·asrc ; t[n] = hW[n]·adst
__global__ void rowdot2_kernel(const float* __restrict__ hW, const float* __restrict__ asrc,
                               const float* __restrict__ adst, float* __restrict__ s,
                               float* __restrict__ t, int N) {
  __shared__ float rs[D];
  __shared__ float rt[D];
  int n = blockIdx.x, d = threadIdx.x;
  float hv = hW[(size_t)n * D + d];
  rs[d] = hv * asrc[d];
  rt[d] = hv * adst[d];
  __syncthreads();
  for (int off = D / 2; off > 0; off >>= 1) {
    if (d < off) { rs[d] += rs[d + off]; rt[d] += rt[d + off]; }
    __syncthreads();
  }
  if (d == 0) { s[n] = rs[0]; t[n] = rt[0]; }
}

// e = leaky_relu(s[src]+t[dst]); segment max into menc[dst]
__global__ void edge_score_kernel(const int* __restrict__ esrc, const int* __restrict__ edst,
                                  const float* __restrict__ s, const float* __restrict__ t,
                                  float* __restrict__ ebuf, unsigned* __restrict__ menc,
                                  int E, int Etot) {
  int e = blockIdx.x * blockDim.x + threadIdx.x;
  if (e >= Etot) return;
  int sn = (e < E) ? esrc[e] : (e - E);   // self-loops appended
  int dn = (e < E) ? edst[e] : (e - E);
  float v = s[sn] + t[dn];
  v = (v > 0.f) ? v : NEG_SLOPE * v;
  ebuf[e] = v;
  atomicMax(&menc[dn], fenc(v));
}

// ebuf = exp(e - m[dst]); denom[dst] += ebuf
__global__ void edge_exp_kernel(const int* __restrict__ edst, const unsigned* __restrict__ menc,
                                float* __restrict__ ebuf, float* __restrict__ denom,
                                int E, int Etot) {
  int e = blockIdx.x * blockDim.x + threadIdx.x;
  if (e >= Etot) return;
  int dn = (e < E) ? edst[e] : (e - E);
  float ee = expf(ebuf[e] - fdec(menc[dn]));
  ebuf[e] = ee;
  atomicAdd(&denom[dn], ee);
}

// agg[dst] += (ebuf[e]/denom[dst]) * hW[src] ; one edge per 256-thread block
__global__ void edge_aggregate_kernel(const int* __restrict__ esrc, const int* __restrict__ edst,
                                      const float* __restrict__ ebuf, const float* __restrict__ denom,
                                      const float* __restrict__ hW, float* __restrict__ agg, int E) {
  int e = blockIdx.x;
  int d = threadIdx.x;
  int sn = (e < E) ? esrc[e] : (e - E);
  int dn = (e < E) ? edst[e] : (e - E);
  float alpha = ebuf[e] / denom[dn];
  atomicAdd(&agg[(size_t)dn * D + d], alpha * hW[(size_t)sn * D + d]);
}

__global__ void bias_relu_kernel(const float* __restrict__ agg, const float* __restrict__ bias,
                                 float* __restrict__ h, int total) {
  int i = blockIdx.x * blockDim.x + threadIdx.x;
  if (i >= total) return;
  float v = agg[i] + bias[i & (D - 1)];
  h[i] = fmaxf(v, 0.f);
}

__global__ void pool_kernel(const float* __restrict__ h, const int* __restrict__ batch,
                            float* __restrict__ padd, unsigned* __restrict__ pmax,
                            float* __restrict__ pcount, int N) {
  int n = blockIdx.x, d = threadIdx.x;
  int b = batch[n];
  float v = h[(size_t)n * D + d];
  atomicAdd(&padd[(size_t)b * D + d], v);
  atomicMax(&pmax[(size_t)b * D + d], fenc(v));
  if (d == 0) atomicAdd(&pcount[b], 1.0f);
}

__global__ void feats_kernel(const float* __restrict__ padd, const unsigned* __restrict__ pmax,
                             const float* __restrict__ pcount, float* __restrict__ feats, int B) {
  int b = blockIdx.x, d = threadIdx.x;
  float cnt = pcount[b];
  float add = padd[(size_t)b * D + d];
  float mx  = (cnt > 0.f) ? fdec(pmax[(size_t)b * D + d]) : 0.f;
  float* fr = feats + (size_t)b * 3 * D;
  fr[d]         = add / fmaxf(cnt, 1.f);
  fr[D + d]     = add;
  fr[2 * D + d] = mx;
}

extern "C" void kernel_launch(void* const* d_in, const int* in_sizes, int n_in,
                              void* d_out, int out_size, void* d_ws, size_t ws_size,
                              hipStream_t stream) {
  const int*   x        = (const int*)d_in[0];
  const int*   eidx     = (const int*)d_in[1];
  const int*   batch    = (const int*)d_in[2];
  const float* atom_emb = (const float*)d_in[3];
  const float* Ws       = (const float*)d_in[4];
  const float* a_src    = (const float*)d_in[5];
  const float* a_dst    = (const float*)d_in[6];
  const float* biases   = (const float*)d_in[7];
  const float* lin_w    = (const float*)d_in[8];
  const float* lin_b    = (const float*)d_in[9];
  float* out = (float*)d_out;

  const int N    = in_sizes[2];
  const int E    = in_sizes[1] / 2;
  const int Etot = E + N;
  const int B    = out_size / OUTF;

  const int* esrc = eidx;
  const int* edst = eidx + E;

  // ---- carve workspace ----
  char* ws = (char*)d_ws;
  float*    h     = (float*)ws;    ws += (size_t)N * D * 4;
  float*    hW    = (float*)ws;    ws += (size_t)N * D * 4;
  float*    agg   = (float*)ws;    ws += (size_t)N * D * 4;
  float*    s     = (float*)ws;    ws += (size_t)N * 4;
  float*    t     = (float*)ws;    ws += (size_t)N * 4;
  unsigned* menc  = (unsigned*)ws; ws += (size_t)N * 4;
  float*    denom = (float*)ws;    ws += (size_t)N * 4;
  float*    ebuf  = (float*)ws;    ws += (size_t)Etot * 4;
  float*    pcnt  = (float*)ws;    ws += (size_t)B * 4;
  float*    padd  = (float*)ws;    ws += (size_t)B * D * 4;
  unsigned* pmax  = (unsigned*)ws; ws += (size_t)B * D * 4;
  float*    feats = (float*)ws;    ws += (size_t)B * 3 * D * 4;
  (void)ws_size; (void)n_in;

  // ---- atom encoder ----
  atom_encoder_kernel<<<N, D, 0, stream>>>(x, atom_emb, h, N);

  const int ND = N * D;
  // ---- GAT layers ----
  for (int l = 0; l < NLAYERS; ++l) {
    fill_u32_kernel<<<(ND + 255) / 256, 256, 0, stream>>>((unsigned*)agg, 0u, ND);
    fill_u32_kernel<<<(N + 255) / 256, 256, 0, stream>>>(menc, ENC_NEG_INF, N);
    fill_u32_kernel<<<(N + 255) / 256, 256, 0, stream>>>((unsigned*)denom, 0u, N);

    dim3 ggrid(N / 16, D / 64);
    wmma_gemm_f32_kernel<<<ggrid, 32, 0, stream>>>(h, Ws + (size_t)l * D * D, nullptr, hW, D, D);

    rowdot2_kernel<<<N, D, 0, stream>>>(hW, a_src + l * D, a_dst + l * D, s, t, N);
    edge_score_kernel<<<(Etot + 255) / 256, 256, 0, stream>>>(esrc, edst, s, t, ebuf, menc, E, Etot);
    edge_exp_kernel<<<(Etot + 255) / 256, 256, 0, stream>>>(edst, menc, ebuf, denom, E, Etot);
    edge_aggregate_kernel<<<Etot, D, 0, stream>>>(esrc, edst, ebuf, denom, hW, agg, E);
    bias_relu_kernel<<<(ND + 255) / 256, 256, 0, stream>>>(agg, biases + l * D, h, ND);
  }

  // ---- pooling ----
  const int BD = B * D;
  fill_u32_kernel<<<(B + 255) / 256, 256, 0, stream>>>((unsigned*)pcnt, 0u, B);
  fill_u32_kernel<<<(BD + 255) / 256, 256, 0, stream>>>((unsigned*)padd, 0u, BD);
  fill_u32_kernel<<<(BD + 255) / 256, 256, 0, stream>>>(pmax, ENC_NEG_INF, BD);
  pool_kernel<<<N, D, 0, stream>>>(h, batch, padd, pmax, pcnt, N);
  feats_kernel<<<B, D, 0, stream>>>(padd, pmax, pcnt, feats, B);

  // ---- final linear [B,3D] @ [3D,OUT] + b ----
  dim3 fgrid(B / 16, OUTF / 64);
  wmma_gemm_f32_kernel<<<fgrid, 32, 0, stream>>>(feats, lin_w, lin_b, out, 3 * D, OUTF);
}